// REM_15607911153781
// MI455X (gfx1250) — compile-verified
//
#include <hip/hip_runtime.h>
#include <hip/hip_bf16.h>
#include <math.h>

// REM generator for MI455X (gfx1250, wave32).
//
// out[c][i][j], c in [0,10), i in [0,2048), j in [0,4096), fp32.
// Value depends only on (c, d=i-j):
//   j >  i          -> 0                      (~75% of elements)
//   d >  200        -> farC(c) in {0,1}       (~20%)
//   0 <= d <= 200   -> tab[c][d] (eye folded into d==0)  (~5%)
//
// Pure streaming-store workload: 335.5 MB / 23.3 TB/s ~= 14.4 us floor.
// One block per (c,i) row: 256 threads x 4 float4 quads = 4096 floats.
// Constant regions are emitted via CDNA5 async LDS->global b128 stores
// (ASYNCcnt path); boundary quads via non-temporal global_store_b128.

#define TRUNC_ROWS 2048
#define T_COLS     4096
#define CAP        200
#define NCH        10

typedef __attribute__((ext_vector_type(4))) float v4f;

__global__ __launch_bounds__(256)
void rem_gen_kernel(const float* __restrict__ eta,
                    const float* __restrict__ nu,
                    const float* __restrict__ theta,
                    float* __restrict__ out)
{
    __shared__ float tab[CAP + 1];    // per-channel band table, eye folded in
    __shared__ v4f   zq;              // 16B of zeros   (async-store source)
    __shared__ v4f   fq;              // 16B far-const  (async-store source)

    const int tid = threadIdx.x;
    const int c   = blockIdx.x >> 11;              // channel 0..9
    const int i   = blockIdx.x & (TRUNC_ROWS - 1); // row     0..2047

    // far-region constant: L0==0 there -> lam^0=1, gam^0*cos0=1, gam^0*sin0=0
    const float farC = (c == 6 || c == 7 || c == 9) ? 0.0f : 1.0f;

    // ---- build the 201-entry band table (threads 0..200 only) ----
    if (tid <= CAP) {
        const float lam = tanhf(eta[0]);
        const float gam = 1.0f / (1.0f + expf(-nu[0]));
        const float th  = theta[0];
        const int   d   = tid;
        const float L   = (float)d;
        float v;
        if (c < 2) {
            v = powf(lam, L);                                         // lam^L
        } else if (c == 2) {
            v = ((d & 7)  == 0) ? powf(lam, (float)(d >> 3)) : 1.0f;  // dilated 8
        } else if (c == 3) {
            v = ((d & 15) == 0) ? powf(lam, (float)(d >> 4)) : 1.0f;  // dilated 16
        } else {
            const float g = powf(gam, L);                             // gam^L (full L)
            if (c == 4 || c == 5)      v = g * cosf(L * th);
            else if (c == 6 || c == 7) v = g * sinf(L * th);
            else if (c == 8)           v = g * cosf(((d & 31) == 0 ? (float)(d >> 5) : 0.0f) * th);
            else                       v = g * sinf(((d & 63) == 0 ? (float)(d >> 6) : 0.0f) * th);
        }
        if (d == 0) v -= 1.0f;   // fold  -eye(2048,4096)  into the diagonal entry
        tab[d] = v;
    }
    if (tid == 0) zq = (v4f){0.0f, 0.0f, 0.0f, 0.0f};
    if (tid == 1) fq = (v4f){farC, farC, farC, farC};
    __syncthreads();

    // raw 32-bit LDS byte offsets of the two async-store source quads
    // (flat LDS addresses carry the aperture only in the high 32 bits)
    const unsigned zoff = (unsigned)(uintptr_t)&zq;
    const unsigned foff = (unsigned)(uintptr_t)&fq;

    v4f* __restrict__ rowq =
        (v4f*)(out + ((size_t)c * TRUNC_ROWS + (size_t)i) * T_COLS);

    #pragma unroll
    for (int k = 0; k < 4; ++k) {
        const int q  = tid + (k << 8);   // quad index 0..1023, wave-contiguous
        const int j0 = q << 2;           // first column of this quad
        v4f* p = rowq + q;

        if (j0 > i) {
            // whole quad above the diagonal -> zeros, async LDS->global B128
            asm volatile("global_store_async_from_lds_b128 %0, %1, off"
                         :: "v"(p), "v"(zoff) : "memory");
        } else if (j0 + (CAP + 3) < i) {
            // whole quad beyond the cap -> per-channel constant, async store
            asm volatile("global_store_async_from_lds_b128 %0, %1, off"
                         :: "v"(p), "v"(foff) : "memory");
        } else {
            // boundary quad: per-element select (only ~52 quads per row)
            v4f v;
            const int d0 = i - (j0 + 0);
            const int d1 = i - (j0 + 1);
            const int d2 = i - (j0 + 2);
            const int d3 = i - (j0 + 3);
            v.x = (d0 < 0) ? 0.0f : ((d0 > CAP) ? farC : tab[d0]);
            v.y = (d1 < 0) ? 0.0f : ((d1 > CAP) ? farC : tab[d1]);
            v.z = (d2 < 0) ? 0.0f : ((d2 > CAP) ? farC : tab[d2]);
            v.w = (d3 < 0) ? 0.0f : ((d3 > CAP) ? farC : tab[d3]);
            __builtin_nontemporal_store(v, p);   // global_store_b128, TH=NT
        }
    }

    // defensive: drain async stores before wave teardown
    asm volatile("s_wait_asynccnt 0x0" ::: "memory");
}

extern "C" void kernel_launch(void* const* d_in, const int* in_sizes, int n_in,
                              void* d_out, int out_size, void* d_ws, size_t ws_size,
                              hipStream_t stream)
{
    (void)in_sizes; (void)n_in; (void)out_size; (void)d_ws; (void)ws_size;
    const float* eta   = (const float*)d_in[0];
    const float* nu    = (const float*)d_in[1];
    const float* theta = (const float*)d_in[2];
    float* out = (float*)d_out;

    const dim3 grid(NCH * TRUNC_ROWS);  // 20480 blocks: one (channel,row) each
    const dim3 block(256);              // 8 wave32s
    hipLaunchKernelGGL(rem_gen_kernel, grid, block, 0, stream,
                       eta, nu, theta, out);
}